// Encoder_84310208020972
// MI455X (gfx1250) — compile-verified
//
#include <hip/hip_runtime.h>

#define VOCAB 32000
#define P_DIM 128
#define B_DIM 64
#define L_DIM 4096

// Tiling: 256 threads/block (8 wave32), each thread owns 4 consecutive l values
// -> LTILE = 1024 l per block. Each block covers PTILE = 16 embedding rows.
#define LTILE 1024
#define PTILE 16

typedef __attribute__((ext_vector_type(4))) float v4f;
typedef __attribute__((ext_vector_type(4))) int   v4i;

__global__ __launch_bounds__(256) void Encoder_gather_gfx1250(
    const int* __restrict__ x,      // (B, L) int32 token ids
    const float* __restrict__ w,    // (P, VOCAB) fp32
    float* __restrict__ out)        // (B, P, L) fp32
{
    const int t  = threadIdx.x;
    const int nL = L_DIM / LTILE;               // 4
    const int b  = blockIdx.x / nL;             // batch index
    const int lt = blockIdx.x % nL;             // l-tile within batch
    const int l0 = lt * LTILE + t * 4;          // first of this thread's 4 l's
    const int p0 = blockIdx.y * PTILE;          // first embedding row of tile

    // One coalesced 128-bit index load; indices live in registers and are
    // reused for all PTILE rows (x traffic ~= 1 MiB total).
    const v4i xi = *reinterpret_cast<const v4i*>(x + (size_t)b * L_DIM + l0);

    // Speculative prefetch of the next l-tile's indices (gfx1250
    // global_prefetch_b8); harmless at the last tile (address stays in-bounds
    // region of the 1 MiB index buffer except final block, where it is
    // speculative and silently dropped on translate fail per ISA).
    __builtin_prefetch(x + (size_t)b * L_DIM + l0 + LTILE, 0, 0);

    const float* wrow = w + (size_t)p0 * VOCAB;
    float* orow = out + ((size_t)b * P_DIM + p0) * L_DIM + l0;

    // 16 rows x 4 gathers: 64 independent 4B loads (L2-resident w_in),
    // 16 coalesced 128-bit non-temporal stores (streaming 128 MiB output).
#pragma unroll
    for (int pp = 0; pp < PTILE; ++pp) {
        v4f v;
        v.x = wrow[xi.x];
        v.y = wrow[xi.y];
        v.z = wrow[xi.z];
        v.w = wrow[xi.w];
        __builtin_nontemporal_store(v, reinterpret_cast<v4f*>(orow));
        wrow += VOCAB;
        orow += L_DIM;
    }
}

extern "C" void kernel_launch(void* const* d_in, const int* in_sizes, int n_in,
                              void* d_out, int out_size, void* d_ws, size_t ws_size,
                              hipStream_t stream) {
    const int*   x = (const int*)d_in[0];    // (B, L)
    const float* w = (const float*)d_in[1];  // (P, VOCAB)
    float*     out = (float*)d_out;          // (B, P, L)

    dim3 grid(B_DIM * (L_DIM / LTILE), P_DIM / PTILE);  // (256, 8)
    dim3 block(256);
    Encoder_gather_gfx1250<<<grid, block, 0, stream>>>(x, w, out);
}